// MambaMixer_48653389529601
// MI455X (gfx1250) — compile-verified
//
#include <hip/hip_runtime.h>
#include <hip/hip_bf16.h>
#include <math.h>

// ---------------------------------------------------------------------------
// Mamba-1.4B forward on MI455X (gfx1250, wave32).
//  * GEMMs: V_WMMA_F32_16X16X4_F32, LDS-tiled. Tiles staged with
//    GLOBAL_LOAD_ASYNC_TO_LDS (ASYNCcnt path): A via b128 copies, B via
//    per-lane b32 copies that transpose into Bs[n][k] for free.
//    Fragments are single ds_load_b64 (A row-major, B transposed in LDS).
//  * Scan: 3-pass chunked scan (C=32 chunks of CL=64): sequential depth
//    2048 -> ~160, occupancy x32.
// ~108 GFLOP vs ~178 MB traffic -> memory/latency bound; fp32 WMMA keeps
// reference accuracy at zero throughput cost.
// ---------------------------------------------------------------------------

typedef __attribute__((ext_vector_type(2))) float v2f;
typedef __attribute__((ext_vector_type(8))) float v8f;

#define TBM 64
#define TBN 128
#define TBK 32
#define LPAD 4          // row stride 36 dwords = 144B: 16B-aligned rows, banks spread

#define SC_C  32        // scan chunks
#define SC_CL 64        // scan chunk length (SC_C * SC_CL == L)

#if defined(__gfx1250__) && \
    __has_builtin(__builtin_amdgcn_global_load_async_to_lds_b128) && \
    __has_builtin(__builtin_amdgcn_global_load_async_to_lds_b32)
#define ASYNC_LDS 1
typedef int v4i32 __attribute__((vector_size(16)));
typedef __attribute__((address_space(1))) v4i32 g_v4i;   // global int4
typedef __attribute__((address_space(3))) v4i32 l_v4i;   // LDS int4
typedef __attribute__((address_space(1))) int   g_i32;   // global int
typedef __attribute__((address_space(3))) int   l_i32;   // LDS int
#else
#define ASYNC_LDS 0
#endif

static __device__ __forceinline__ void wait_async0() {
#if defined(__gfx1250__)
#if __has_builtin(__builtin_amdgcn_s_wait_asynccnt)
    __builtin_amdgcn_s_wait_asynccnt(0);
#else
    asm volatile("s_wait_asynccnt 0x0" ::: "memory");
#endif
#endif
}

// Generic row-major f32 GEMM: C[M,N] (+epilogue) = A[M,K] @ B[K,N]
// epilogue: 0 = plain store, 1 = softplus(x + bias[n])
__global__ __launch_bounds__(256) void gemm_f32_wmma(
    const float* __restrict__ A, int lda,
    const float* __restrict__ B, int ldb,
    float* __restrict__ C, int ldc,
    int M, int N, int K,
    const float* __restrict__ bias, int epilogue)
{
    __shared__ __align__(16) float As[TBM][TBK + LPAD];   // [m][k]
    __shared__ __align__(16) float Bs[TBN][TBK + LPAD];   // [n][k]  (transposed)

    const int tid  = threadIdx.x;
    const int wave = tid >> 5;           // 0..7
    const int lane = tid & 31;
    const int wm   = wave >> 2;          // 0..1  (wave row in 2x4 wave grid)
    const int wn   = wave & 3;           // 0..3  (wave col)
    const int block_m = blockIdx.y * TBM;
    const int block_n = blockIdx.x * TBN;

    const int mn_in_lane = lane & 15;        // row (A) / col (B,C)
    const int khalf      = (lane >> 4) * 2;  // 0 or 2: K-pair owned by half-wave
    const int row_half   = (lane >> 4) * 8;  // C rows v or v+8

    v8f acc[2][2];
#pragma unroll
    for (int im = 0; im < 2; ++im)
#pragma unroll
        for (int in = 0; in < 2; ++in)
            acc[im][in] = (v8f){0.f, 0.f, 0.f, 0.f, 0.f, 0.f, 0.f, 0.f};

    for (int k0 = 0; k0 < K; k0 += TBK) {
#if ASYNC_LDS
        const bool a_fast = (block_m + TBM <= M) && (k0 + TBK <= K);
        const bool b_fast = (block_n + TBN <= N) && (k0 + TBK <= K);
#else
        const bool a_fast = false, b_fast = false;
#endif

#if ASYNC_LDS
        if (a_fast) {
            // A tile: 64x32 f32 = 512 x 16B; 2 async b128 copies per thread.
#pragma unroll
            for (int j = 0; j < 2; ++j) {
                int t = tid + j * 256;          // 0..511
                int r = t >> 3;                 // m-in-tile
                int c = (t & 7) * 4;            // k-in-tile (16B aligned)
                const float* gp = A + (size_t)(block_m + r) * lda + (k0 + c);
                __builtin_amdgcn_global_load_async_to_lds_b128(
                    (g_v4i*)gp, (l_v4i*)&As[r][c], 0, 0);
            }
        } else
#endif
        {
            for (int t = tid; t < TBM * TBK; t += 256) {
                int r = t / TBK, c = t % TBK;
                int gm = block_m + r, gk = k0 + c;
                As[r][c] = (gm < M && gk < K) ? A[(size_t)gm * lda + gk] : 0.f;
            }
        }

#if ASYNC_LDS
        if (b_fast) {
            // B tile: 32x128 f32; per-lane b32 async copies transpose into Bs[n][k].
#pragma unroll
            for (int j = 0; j < (TBK * TBN) / 256; ++j) {
                int t = tid + j * 256;          // 0..4095
                int r = t >> 7;                 // k-in-tile
                int c = t & 127;                // n-in-tile (coalesced global reads)
                const float* gp = B + (size_t)(k0 + r) * ldb + (block_n + c);
                __builtin_amdgcn_global_load_async_to_lds_b32(
                    (g_i32*)gp, (l_i32*)&Bs[c][r], 0, 0);
            }
        } else
#endif
        {
            for (int t = tid; t < TBK * TBN; t += 256) {
                int r = t / TBN, c = t % TBN;   // r = k, c = n
                int gk = k0 + r, gn = block_n + c;
                Bs[c][r] = (gk < K && gn < N) ? B[(size_t)gk * ldb + gn] : 0.f;
            }
        }

#if ASYNC_LDS
        if (a_fast || b_fast) wait_async0();
#endif
        __syncthreads();

#pragma unroll
        for (int kk = 0; kk < TBK; kk += 4) {
            v2f afrag[2], bfrag[2];
#pragma unroll
            for (int im = 0; im < 2; ++im) {
                int r = wm * 32 + im * 16 + mn_in_lane;
                afrag[im] = *(const v2f*)&As[r][kk + khalf];      // ds_load_b64
            }
#pragma unroll
            for (int in = 0; in < 2; ++in) {
                int c = wn * 32 + in * 16 + mn_in_lane;
                bfrag[in] = *(const v2f*)&Bs[c][kk + khalf];      // ds_load_b64
            }
#pragma unroll
            for (int im = 0; im < 2; ++im)
#pragma unroll
                for (int in = 0; in < 2; ++in)
                    acc[im][in] = __builtin_amdgcn_wmma_f32_16x16x4_f32(
                        false, afrag[im], false, bfrag[in],
                        (short)0, acc[im][in], false, false);
        }
        __syncthreads();
    }

#pragma unroll
    for (int im = 0; im < 2; ++im)
#pragma unroll
        for (int in = 0; in < 2; ++in)
#pragma unroll
            for (int v = 0; v < 8; ++v) {
                int gm = block_m + wm * 32 + im * 16 + row_half + v;
                int gn = block_n + wn * 32 + in * 16 + mn_in_lane;
                if (gm < M && gn < N) {
                    float val = acc[im][in][v];
                    if (epilogue == 1) {
                        val += bias[gn];
                        val = (val > 20.f) ? val : log1pf(__expf(val));
                    }
                    C[(size_t)gm * ldc + gn] = val;
                }
            }
}

// Depthwise causal conv (K=4) over seq + bias + SiLU.
__global__ void conv_silu_kernel(const float* __restrict__ proj,
                                 const float* __restrict__ w,
                                 const float* __restrict__ b,
                                 float* __restrict__ u,
                                 int L, int I, int twoI)
{
    int idx = blockIdx.x * blockDim.x + threadIdx.x;
    if (idx >= L * I) return;
    int l = idx / I, i = idx - l * I;
    float acc = b[i];
#pragma unroll
    for (int k = 0; k < 4; ++k) {
        int ls = l - 3 + k;
        if (ls >= 0) acc += proj[(size_t)ls * twoI + i] * w[i * 4 + k];
    }
    u[idx] = acc / (1.f + __expf(-acc));   // silu
}

// ---- Chunked selective scan --------------------------------------------
// Recurrence per (i,n): s_l = s_{l-1} * exp(dt_l * A) + dt_l * u_l * B_l
// Pass A: per chunk, local state (from 0) and decay product.
__global__ __launch_bounds__(256) void scan_chunk_local(
    const float* __restrict__ dt, const float* __restrict__ u,
    const float* __restrict__ ssm_in, const float* __restrict__ A_log,
    float* __restrict__ Pbuf, float* __restrict__ Sbuf, int I)
{
    const size_t idx = (size_t)blockIdx.x * 256 + threadIdx.x;
    const int n = (int)(idx & 15);
    const int i = (int)((idx >> 4) & (size_t)(I - 1));
    const int c = (int)(idx >> 16);              // (idx>>4)/I with I=4096
    const float Ai = -__expf(A_log[i * 16 + n]);

    float s = 0.f, p = 1.f;
    const int l0 = c * SC_CL;
    for (int dl = 0; dl < SC_CL; ++dl) {
        int l = l0 + dl;
        float dtv = dt[(size_t)l * I + i];
        float uv  = u[(size_t)l * I + i];
        float Bv  = ssm_in[(size_t)l * 160 + 128 + n];
        float e   = __expf(dtv * Ai);
        s = s * e + dtv * uv * Bv;
        p *= e;
    }
    const size_t o = (size_t)c * ((size_t)I * 16) + (size_t)i * 16 + n;
    Pbuf[o] = p;
    Sbuf[o] = s;
}

// Pass B: sequential prefix over chunk summaries; Sbuf[c] <- incoming state.
__global__ __launch_bounds__(256) void scan_chunk_prefix(
    float* __restrict__ Pbuf, float* __restrict__ Sbuf, int IN /* I*16 */)
{
    const int j = blockIdx.x * 256 + threadIdx.x;
    if (j >= IN) return;
    float carry = 0.f;
    for (int c = 0; c < SC_C; ++c) {
        size_t o = (size_t)c * IN + j;
        float p = Pbuf[o];
        float s = Sbuf[o];
        Sbuf[o] = carry;              // state entering chunk c
        carry = carry * p + s;
    }
}

// Pass C: replay each chunk from its incoming state, emit y.
__global__ __launch_bounds__(256) void scan_chunk_emit(
    const float* __restrict__ dt, const float* __restrict__ u,
    const float* __restrict__ ssm_in, const float* __restrict__ A_log,
    const float* __restrict__ Sbuf, float* __restrict__ y, int I)
{
    const size_t idx = (size_t)blockIdx.x * 256 + threadIdx.x;
    const int n = (int)(idx & 15);
    const int i = (int)((idx >> 4) & (size_t)(I - 1));
    const int c = (int)(idx >> 16);
    const float Ai = -__expf(A_log[i * 16 + n]);

    float s = Sbuf[(size_t)c * ((size_t)I * 16) + (size_t)i * 16 + n];
    const int l0 = c * SC_CL;
    for (int dl = 0; dl < SC_CL; ++dl) {
        int l = l0 + dl;
        float dtv = dt[(size_t)l * I + i];
        float uv  = u[(size_t)l * I + i];
        float Bv  = ssm_in[(size_t)l * 160 + 128 + n];
        float Cv  = ssm_in[(size_t)l * 160 + 144 + n];
        s = s * __expf(dtv * Ai) + dtv * uv * Bv;
        float pr = s * Cv;
        pr += __shfl_xor(pr, 1, 16);          // 16-lane butterfly reduce
        pr += __shfl_xor(pr, 2, 16);
        pr += __shfl_xor(pr, 4, 16);
        pr += __shfl_xor(pr, 8, 16);
        if (n == 0) y[(size_t)l * I + i] = pr;
    }
}

// y2 = (y + u*D) * silu(gate),  gate = proj[:, I:2I]
__global__ void gate_kernel(const float* __restrict__ y,
                            const float* __restrict__ u,
                            const float* __restrict__ D,
                            const float* __restrict__ proj,
                            float* __restrict__ y2,
                            int L, int I, int twoI)
{
    int idx = blockIdx.x * blockDim.x + threadIdx.x;
    if (idx >= L * I) return;
    int l = idx / I, i = idx - l * I;
    float g = proj[(size_t)l * twoI + I + i];
    y2[idx] = (y[idx] + u[idx] * D[i]) * (g / (1.f + __expf(-g)));
}

extern "C" void kernel_launch(void* const* d_in, const int* in_sizes, int n_in,
                              void* d_out, int out_size, void* d_ws, size_t ws_size,
                              hipStream_t stream) {
    (void)in_sizes; (void)n_in; (void)out_size; (void)ws_size;

    const float* x          = (const float*)d_in[0];  // (1, L, H)
    const float* in_proj_w  = (const float*)d_in[1];  // (H, 2I)
    const float* conv_w     = (const float*)d_in[2];  // (I, 1, 4)
    const float* conv_b     = (const float*)d_in[3];  // (I,)
    const float* x_proj_w   = (const float*)d_in[4];  // (I, 160)
    const float* dt_proj_w  = (const float*)d_in[5];  // (128, I)
    const float* dt_proj_b  = (const float*)d_in[6];  // (I,)
    const float* A_log      = (const float*)d_in[7];  // (I, 16)
    const float* Dvec       = (const float*)d_in[8];  // (I,)
    const float* out_proj_w = (const float*)d_in[9];  // (I, H)
    float* out = (float*)d_out;                       // (1, L, H)

    const int L = 2048, H = 2048, I = 4096, twoI = 8192, R = 128, RN = 160;
    const int IN = I * 16;

    float* ws   = (float*)d_ws;
    float* proj = ws;                          // L * 2I   (64 MB)
    float* u    = proj + (size_t)L * twoI;     // L * I    (32 MB)
    float* ssm  = u    + (size_t)L * I;        // L * 160  (1.25 MB)
    float* dt   = ssm  + (size_t)L * RN;       // L * I    (32 MB)
    float* yb   = dt   + (size_t)L * I;        // L * I    (32 MB)
    float* Pbuf = yb   + (size_t)L * I;        // C * I*16 (8.4 MB)
    float* Sbuf = Pbuf + (size_t)SC_C * IN;    // C * I*16 (8.4 MB)
    float* y2   = dt;  // scan fully consumes dt before gate_kernel writes here

    // 1) proj = x @ in_proj_w                 (L x H)(H x 2I)
    gemm_f32_wmma<<<dim3(twoI / TBN, L / TBM), 256, 0, stream>>>(
        x, H, in_proj_w, twoI, proj, twoI, L, twoI, H, nullptr, 0);

    // 2) u = silu(depthwise_causal_conv(hidden) + b)
    conv_silu_kernel<<<(L * I + 255) / 256, 256, 0, stream>>>(
        proj, conv_w, conv_b, u, L, I, twoI);

    // 3) ssm_in = u @ x_proj_w               (L x I)(I x 160)
    gemm_f32_wmma<<<dim3((RN + TBN - 1) / TBN, L / TBM), 256, 0, stream>>>(
        u, I, x_proj_w, RN, ssm, RN, L, RN, I, nullptr, 0);

    // 4) dt = softplus(ssm_in[:, :128] @ dt_proj_w + dt_proj_b)  (A stride 160)
    gemm_f32_wmma<<<dim3(I / TBN, L / TBM), 256, 0, stream>>>(
        ssm, RN, dt_proj_w, I, dt, I, L, I, R, dt_proj_b, 1);

    // 5) chunked selective scan -> yb
    {
        int scan_threads = SC_C * IN;                       // 2M
        scan_chunk_local<<<scan_threads / 256, 256, 0, stream>>>(
            dt, u, ssm, A_log, Pbuf, Sbuf, I);
        scan_chunk_prefix<<<(IN + 255) / 256, 256, 0, stream>>>(Pbuf, Sbuf, IN);
        scan_chunk_emit<<<scan_threads / 256, 256, 0, stream>>>(
            dt, u, ssm, A_log, Sbuf, yb, I);
    }

    // 6) y2 = (y + u*D) * silu(gate)
    gate_kernel<<<(L * I + 255) / 256, 256, 0, stream>>>(
        yb, u, Dvec, proj, y2, L, I, twoI);

    // 7) out = y2 @ out_proj_w               (L x I)(I x H)
    gemm_f32_wmma<<<dim3(H / TBN, L / TBM), 256, 0, stream>>>(
        y2, I, out_proj_w, H, out, H, L, H, I, nullptr, 0);
}